// DAP_72189810312075
// MI455X (gfx1250) — compile-verified
//
#include <hip/hip_runtime.h>

typedef __attribute__((ext_vector_type(2))) float v2f;
typedef __attribute__((ext_vector_type(8))) float v8f;

#define B_    8
#define C_IN  16
#define H_    256
#define W_    256
#define NCH   9          // conv channels actually consumed by the shift-average
#define KDIM  144        // 16 in-ch * 3 * 3
#define NPIX  (B_ * H_ * W_)   // 524288
#define BN_EPS 1e-5f

#define WSTRIDE 148      // padded K-stride for weight LDS (conflict-free, even)
#define XT_W    132      // padded row width for x tile LDS (128 pixels + 2 halo + pad)

#if defined(__has_builtin)
# if __has_builtin(__builtin_amdgcn_global_load_async_to_lds_b32)
#  define HAVE_ASYNC_LDS 1
# endif
#endif

typedef __attribute__((address_space(1))) int gi_t;   // global int
typedef __attribute__((address_space(3))) int li_t;   // LDS int

// LDS offset of x-tile element for flattened K index k = ci*9 + kh*3 + kw.
// Pure function of k -> constant-folds after full unroll.
__device__ __forceinline__ constexpr int xtoff(int k) {
    return ((k / 9) * 3 + (k % 9) / 3) * XT_W + (k % 9) % 3;
}

// -------- workspace layout (floats) --------
// [0, NCH*NPIX)            : conv+bias output planes y[c][b][h][w]
// [NCH*NPIX + 0  .. +15]   : per-channel sum
// [NCH*NPIX + 16 .. +31]   : per-channel sumsq
// [NCH*NPIX + 32 .. +47]   : BN scale  a_c = gamma*rsqrt(var+eps)
// [NCH*NPIX + 48 .. +63]   : BN shift  b_c = beta - mu*a_c

// ---------------------------------------------------------------------------
// Pass 1: implicit-GEMM 3x3 conv via V_WMMA_F32_16X16X4_F32 + BN statistics.
// Block = 8 waves = 128 contiguous pixels of one image row.
// x tile (3 rows x 130 cols x 16 ch, halo pre-zeroed) staged in LDS via
// async global->LDS loads; K-loop is branch-free ds_load + wmma.
// ---------------------------------------------------------------------------
__global__ __launch_bounds__(256) void conv_wmma_stats(
    const float* __restrict__ x, const float* __restrict__ wgt,
    const float* __restrict__ bias, float* __restrict__ yws,
    float* __restrict__ gstats)
{
    __shared__ float Wlds[16 * WSTRIDE];        // Wlds[m*WSTRIDE + k]
    __shared__ float xt[C_IN * 3 * XT_W];       // xt[(ci*3+kh)*XT_W + l]
    __shared__ float biasLds[16];
    __shared__ float sSum[NCH];
    __shared__ float sSq[NCH];

    const int tid = threadIdx.x;

    const int blk  = blockIdx.x;                // 4096 blocks
    const int wseg = blk & 1;                   // 2 segments of 128 pixels per row
    const int h    = (blk >> 1) & 255;
    const int b    = blk >> 9;

    // ---- stage weights: Wlds[m][k] = wgt[m*144+k] for m<9, zero-padded ----
    for (int i = tid; i < 16 * KDIM; i += 256) {
        int m = i / KDIM, k = i - m * KDIM;
        Wlds[m * WSTRIDE + k] = (m < NCH) ? wgt[i] : 0.0f;
    }
    if (tid < 16) biasLds[tid] = (tid < NCH) ? bias[tid] : 0.0f;
    if (tid < NCH) { sSum[tid] = 0.0f; sSq[tid] = 0.0f; }

    // ---- stage x tile (with zeroed halo) ----
    const float* xb = x + (size_t)b * (C_IN * H_ * W_);
    const int NXT = C_IN * 3 * 130;             // 6240 elements
    for (int i = tid; i < NXT; i += 256) {
        int ci = i / 390;
        int r  = i - ci * 390;
        int kh = r / 130;
        int l  = r - kh * 130;
        int hy = h + kh - 1;
        int wx = wseg * 128 - 1 + l;
        int lidx = (ci * 3 + kh) * XT_W + l;
        bool ok = (hy >= 0) & (hy < H_) & (wx >= 0) & (wx < W_);
#if HAVE_ASYNC_LDS
        if (ok) {
            __builtin_amdgcn_global_load_async_to_lds_b32(
                (gi_t*)(xb + (ci * H_ + hy) * W_ + wx), (li_t*)&xt[lidx], 0, 0);
        } else {
            xt[lidx] = 0.0f;
        }
#else
        xt[lidx] = ok ? xb[(ci * H_ + hy) * W_ + wx] : 0.0f;
#endif
    }
#if HAVE_ASYNC_LDS
    asm volatile("s_wait_asynccnt 0x0" ::: "memory");
#endif
    __syncthreads();

    const int wave = tid >> 5;
    const int lane = tid & 31;
    const int n    = lane & 15;                 // M row (A) / N column (B)
    const bool hi16 = (lane >> 4) != 0;         // half-wave select
    const int koff = hi16 ? 2 : 0;              // K sub-offset per ISA A/B layout
    const int base_l = wave * 16 + n;           // local pixel column in x tile
    const int awbase = n * WSTRIDE + koff;      // A fetch base (k0 is imm offset)

    // ---- K loop: ds_load_b64(A) + 2x ds_load_b32(B) + wmma, fully unrolled;
    //      B addresses are (base + cndmask-of-two-constants) after folding ----
    v8f acc = {};
#pragma unroll
    for (int k0 = 0; k0 < KDIM; k0 += 4) {
        v2f av = *(const v2f*)&Wlds[awbase + k0];        // A[m][ka], A[m][ka+1]
        int i0 = base_l + (hi16 ? xtoff(k0 + 2) : xtoff(k0));
        int i1 = base_l + (hi16 ? xtoff(k0 + 3) : xtoff(k0 + 1));
        v2f bv; bv.x = xt[i0]; bv.y = xt[i1];
        acc = __builtin_amdgcn_wmma_f32_16x16x4_f32(
            /*neg_a=*/false, av, /*neg_b=*/false, bv,
            /*c_mod=*/(short)0, acc, /*reuse_a=*/false, /*reuse_b=*/false);
    }

    // C/D layout: lanes 0-15 -> M = vgpr, lanes 16-31 -> M = vgpr + 8
    const int wpix = wseg * 128 + wave * 16 + n;
    const int pix  = (b * H_ + h) * W_ + wpix;
#pragma unroll
    for (int v = 0; v < 8; ++v) {
        const int ch = hi16 ? (v + 8) : v;
        const bool live = (ch < NCH);
        float val = live ? (acc[v] + biasLds[ch]) : 0.0f;
        if (live) yws[(size_t)ch * NPIX + pix] = val;

        // reduce over the 16 lanes that share this channel
        float r = val;
        float q = val * val;
        r += __shfl_xor(r, 1);  q += __shfl_xor(q, 1);
        r += __shfl_xor(r, 2);  q += __shfl_xor(q, 2);
        r += __shfl_xor(r, 4);  q += __shfl_xor(q, 4);
        r += __shfl_xor(r, 8);  q += __shfl_xor(q, 8);
        if (live && (lane & 15) == 0) {
            atomicAdd(&sSum[ch], r);   // ds_add_f32
            atomicAdd(&sSq[ch], q);
        }
    }
    __syncthreads();
    if (tid < NCH) {
        unsafeAtomicAdd(&gstats[tid],      sSum[tid]);   // global_atomic_add_f32
        unsafeAtomicAdd(&gstats[16 + tid], sSq[tid]);
    }
}

// ---------------------------------------------------------------------------
// Pass 2: fold BN into per-channel affine (a, b).
// ---------------------------------------------------------------------------
__global__ void bn_coef(const float* __restrict__ gamma,
                        const float* __restrict__ beta,
                        float* __restrict__ gstats)
{
    int c = threadIdx.x;
    if (c < NCH) {
        const float invN = 1.0f / (float)NPIX;
        float mu  = gstats[c] * invN;
        float var = gstats[16 + c] * invN - mu * mu;     // biased, like torch BN fwd
        float rs  = rsqrtf(var + BN_EPS);
        float a   = gamma[c] * rs;
        gstats[32 + c] = a;
        gstats[48 + c] = beta[c] - mu * a;
    }
}

// ---------------------------------------------------------------------------
// Pass 3: BN + ReLU applied on the fly, 9-way shift average (zero-padded),
// broadcast to all 16 output channels.
// ---------------------------------------------------------------------------
__global__ __launch_bounds__(256) void shift_avg(
    const float* __restrict__ yws, const float* __restrict__ coef,
    float* __restrict__ out)
{
    const int p = blockIdx.x * 256 + threadIdx.x;   // pixel id in [0, NPIX)
    const int b = p >> 16;
    const int hw = p & 65535;
    const int h = hw >> 8;
    const int w = hw & 255;

    float s = 0.0f;
#pragma unroll
    for (int c = 0; c < NCH; ++c) {
        const int dy = c / 3 - 1, dx = c % 3 - 1;   // compile-time per iteration
        const int hy = h + dy, wx = w + dx;
        if (hy >= 0 && hy < H_ && wx >= 0 && wx < W_) {
            float v = yws[(size_t)c * NPIX + (b << 16) + hy * W_ + wx];
            float t = fmaf(coef[32 + c], v, coef[48 + c]);
            s += fmaxf(t, 0.0f);                    // ReLU before zero-pad shift
        }
    }
    const float avg = s * (1.0f / 9.0f);

    const size_t obase = ((size_t)b * C_IN) << 16;
    const int hwoff = h * W_ + w;
#pragma unroll
    for (int co = 0; co < C_IN; ++co)
        out[obase + ((size_t)co << 16) + hwoff] = avg;
}

// ---------------------------------------------------------------------------
extern "C" void kernel_launch(void* const* d_in, const int* in_sizes, int n_in,
                              void* d_out, int out_size, void* d_ws, size_t ws_size,
                              hipStream_t stream)
{
    const float* x     = (const float*)d_in[0];
    const float* wgt   = (const float*)d_in[1];
    const float* bias  = (const float*)d_in[2];
    const float* gamma = (const float*)d_in[3];
    const float* beta  = (const float*)d_in[4];
    float* out = (float*)d_out;

    float* yws    = (float*)d_ws;                       // 9 * 524288 f32 = 18.9 MB
    float* gstats = yws + (size_t)NCH * NPIX;           // 64 f32 of stats/coefs

    (void)hipMemsetAsync(gstats, 0, 64 * sizeof(float), stream);  // deterministic

    conv_wmma_stats<<<B_ * H_ * 2, 256, 0, stream>>>(x, wgt, bias, yws, gstats);
    bn_coef<<<1, 32, 0, stream>>>(gamma, beta, gstats);
    shift_avg<<<NPIX / 256, 256, 0, stream>>>(yws, gstats, out);
}